// GraphModel_22832046145851
// MI455X (gfx1250) — compile-verified
//
#include <hip/hip_runtime.h>
#include <hip/hip_bf16.h>

typedef __attribute__((ext_vector_type(2))) float v2f;
typedef __attribute__((ext_vector_type(8))) float v8f;

constexpr int Nn   = 50000;
constexpr int Ee   = 1600000;
constexpr int DIN  = 256;
constexpr int Hh   = 128;
constexpr int DOUT = 64;
constexpr int MT   = Nn / 16;   // 3125 row tiles (exact)

// ---------------- degree / dinv ----------------
__global__ void fill_ones(float* __restrict__ p, int n) {
    int i = blockIdx.x * blockDim.x + threadIdx.x;
    if (i < n) p[i] = 1.0f;                      // self-loop contribution
}

__global__ void deg_edges(const int* __restrict__ idx, float* __restrict__ deg, int e) {
    int i = blockIdx.x * blockDim.x + threadIdx.x;
    if (i < e) atomicAdd(&deg[idx[e + i]], 1.0f);   // dst = idx[E + i]
}

__global__ void make_dinv(float* __restrict__ p, int n) {
    int i = blockIdx.x * blockDim.x + threadIdx.x;
    if (i < n) { float d = p[i]; p[i] = (d > 0.0f) ? rsqrtf(d) : 0.0f; }
}

// ---------------- GEMM: C[M x F] = (relu?)A[M x K] @ W[K x F] --------------
// One wave -> 16 rows x all F columns. V_WMMA_F32_16X16X4_F32, K stepped by 4.
template <int K, int F, bool RELU>
__global__ __launch_bounds__(128)
void gemm_wmma(const float* __restrict__ A, const float* __restrict__ W,
               float* __restrict__ C, int mtiles) {
    const int wave = blockIdx.x * (blockDim.x >> 5) + (threadIdx.x >> 5);
    if (wave >= mtiles) return;
    const int lane = threadIdx.x & 31;
    const int half = lane >> 4;      // 0: K0/K1, 1: K2/K3 (A & B fragment halves)
    const int l16  = lane & 15;

    constexpr int NT = F / 16;
    v8f acc[NT] = {};                // zero-init accumulators

    const float* Arow = A + (size_t)(wave * 16 + l16) * K + 2 * half;
    for (int k0 = 0; k0 < K; k0 += 4) {
        float2 af = *reinterpret_cast<const float2*>(Arow + k0);
        if (RELU) { af.x = fmaxf(af.x, 0.0f); af.y = fmaxf(af.y, 0.0f); }
        v2f a; a.x = af.x; a.y = af.y;
        const float* Wk = W + (size_t)(k0 + 2 * half) * F + l16;
#pragma unroll
        for (int t = 0; t < NT; ++t) {
            v2f b; b.x = Wk[t * 16]; b.y = Wk[t * 16 + F];
            acc[t] = __builtin_amdgcn_wmma_f32_16x16x4_f32(
                false, a, false, b, (short)0, acc[t], false, false);
        }
    }
    // C/D layout: VGPR r -> row (r + 8*half), lane&15 -> column
#pragma unroll
    for (int t = 0; t < NT; ++t) {
#pragma unroll
        for (int r = 0; r < 8; ++r) {
            int row = wave * 16 + r + 8 * half;
            C[(size_t)row * F + t * 16 + l16] = acc[t][r];
        }
    }
}

// ------------- aggregation init: dest = bias + dinv[n]^2 * h[n] -------------
template <int F>
__global__ void agg_init(const float* __restrict__ h, const float* __restrict__ dinv,
                         const float* __restrict__ bias, float* __restrict__ dest, int n) {
    int i = blockIdx.x * blockDim.x + threadIdx.x;
    if (i >= n * F) return;
    int node = i / F, f = i % F;
    float di = dinv[node];
    dest[i] = bias[f] + di * di * h[i];
}

// ------------- edge scatter: dest[d] += dinv[s]*dinv[d]*h[s] ----------------
// One wave per edge; lane covers F/32 consecutive floats (coalesced gather).
template <int F>
__global__ __launch_bounds__(256)
void agg_scatter(const float* __restrict__ h, const float* __restrict__ dinv,
                 const int* __restrict__ idx, float* __restrict__ dest, int e) {
    constexpr int VEC = F / 32;
    int edge = blockIdx.x * 8 + (threadIdx.x >> 5);
    if (edge >= e) return;
    int lane = threadIdx.x & 31;
    int s = idx[edge];
    int d = idx[e + edge];
    float nrm = dinv[s] * dinv[d];
    const float* hs = h    + (size_t)s * F + lane * VEC;
    float*       dd = dest + (size_t)d * F + lane * VEC;
#pragma unroll
    for (int v = 0; v < VEC; ++v) atomicAdd(&dd[v], nrm * hs[v]);
}

// ---------------------------------------------------------------------------
extern "C" void kernel_launch(void* const* d_in, const int* in_sizes, int n_in,
                              void* d_out, int out_size, void* d_ws, size_t ws_size,
                              hipStream_t stream) {
    const float* x  = (const float*)d_in[0];
    const float* W1 = (const float*)d_in[1];
    const float* b1 = (const float*)d_in[2];
    const float* W2 = (const float*)d_in[3];
    const float* b2 = (const float*)d_in[4];
    const float* W3 = (const float*)d_in[5];
    const float* b3 = (const float*)d_in[6];
    const int*  idx = (const int*)  d_in[7];
    float* out = (float*)d_out;

    char* ws = (char*)d_ws;
    const size_t o_dinv = 0;
    const size_t o_t0   = ((size_t)Nn * 4 + 255) & ~(size_t)255;       // after dinv
    const size_t o_t1   = o_t0 + (size_t)Nn * Hh * 4;
    float* dinv = (float*)(ws + o_dinv);
    float* t0   = (float*)(ws + o_t0);
    float* t1   = (float*)(ws + o_t1);

    // degrees -> dinv
    fill_ones<<<(Nn + 255) / 256, 256, 0, stream>>>(dinv, Nn);
    deg_edges<<<(Ee + 255) / 256, 256, 0, stream>>>(idx, dinv, Ee);
    make_dinv<<<(Nn + 255) / 256, 256, 0, stream>>>(dinv, Nn);

    const int gemmBlocks = (MT + 3) / 4;  // 4 waves/block

    // Layer 1: t0 = x@W1 ; t1 = agg(t0)+b1   (relu fused into next gemm load)
    gemm_wmma<DIN, Hh, false><<<gemmBlocks, 128, 0, stream>>>(x, W1, t0, MT);
    agg_init<Hh><<<(Nn * Hh + 255) / 256, 256, 0, stream>>>(t0, dinv, b1, t1, Nn);
    agg_scatter<Hh><<<(Ee + 7) / 8, 256, 0, stream>>>(t0, dinv, idx, t1, Ee);

    // Layer 2: t0 = relu(t1)@W2 ; t1 = agg(t0)+b2
    gemm_wmma<Hh, Hh, true><<<gemmBlocks, 128, 0, stream>>>(t1, W2, t0, MT);
    agg_init<Hh><<<(Nn * Hh + 255) / 256, 256, 0, stream>>>(t0, dinv, b2, t1, Nn);
    agg_scatter<Hh><<<(Ee + 7) / 8, 256, 0, stream>>>(t0, dinv, idx, t1, Ee);

    // Layer 3: t0 = relu(t1)@W3 ; out = agg(t0)+b3
    gemm_wmma<Hh, DOUT, true><<<gemmBlocks, 128, 0, stream>>>(t1, W3, t0, MT);
    agg_init<DOUT><<<(Nn * DOUT + 255) / 256, 256, 0, stream>>>(t0, dinv, b3, out, Nn);
    agg_scatter<DOUT><<<(Ee + 7) / 8, 256, 0, stream>>>(t0, dinv, idx, out, Ee);
}